// NNLoss_37512244363425
// MI455X (gfx1250) — compile-verified
//
#include <hip/hip_runtime.h>
#include <hip/hip_bf16.h>

typedef __attribute__((ext_vector_type(2))) float v2f;
typedef __attribute__((ext_vector_type(8))) float v8f;

#define NQ 14400     // query patches
#define NK 14400     // key patches
#define DR 75        // real patch dim (3*1*5*5)
#define DP 80        // padded dim (20 WMMA K-steps of 4)
#define TDIM 4
#define HIMG 64
#define WIMG 64

// ---------------------------------------------------------------------------
// Patch extraction: [1,3,4,64,64] -> [14400, 80] padded row-major patches.
// n = t*3600 + y*60 + x ; j = c*25 + dy*5 + dx (order only needs consistency).
// ---------------------------------------------------------------------------
__global__ __launch_bounds__(256) void patch_kernel(const float* __restrict__ vid,
                                                    float* __restrict__ out) {
    int gid = blockIdx.x * blockDim.x + threadIdx.x;
    if (gid >= NQ * DP) return;
    int n = gid / DP;
    int j = gid - n * DP;
    float v = 0.0f;
    if (j < DR) {
        int c  = j / 25;
        int r  = j - c * 25;
        int dy = r / 5;
        int dx = r - dy * 5;
        int t  = n / 3600;
        int r2 = n - t * 3600;
        int y  = r2 / 60;
        int x  = r2 - y * 60;
        v = vid[((c * TDIM + t) * HIMG + (y + dy)) * WIMG + (x + dx)];
    }
    out[n * DP + j] = v;
}

// ---------------------------------------------------------------------------
// Row norms of the key matrix: kn[n] = sum_j k[n][j]^2
// ---------------------------------------------------------------------------
__global__ __launch_bounds__(256) void knorm_kernel(const float* __restrict__ kp,
                                                    float* __restrict__ kn) {
    int n = blockIdx.x * blockDim.x + threadIdx.x;
    if (n >= NK) return;
    const float* row = kp + n * DP;
    float s = 0.0f;
    #pragma unroll 5
    for (int j = 0; j < DR; ++j) s += row[j] * row[j];
    kn[n] = s;
}

// ---------------------------------------------------------------------------
// Exact 1-NN argmin via fp32 WMMA (V_WMMA_F32_16X16X4_F32).
// Block = 128 threads = 4 waves. Each wave owns 3 query tiles (48 queries)
// and sweeps all 900 key tiles of 16, with 3 accumulators for WMMA ILP.
// argmin criterion: d2 = kn[n] - 2*(q . k)   (qn is constant per row).
// Per-lane key stream is strictly increasing -> strict '<' == first-occurrence
// argmin; cross-lane reduce tie-breaks on smaller index.
// ---------------------------------------------------------------------------
#define WAVES 4
#define TQ 3                      // query tiles per wave
#define QPB (WAVES * TQ * 16)     // 192 queries per block

__global__ __launch_bounds__(128) void argmin_kernel(const float* __restrict__ qp,
                                                     const float* __restrict__ kp,
                                                     const float* __restrict__ kn,
                                                     int* __restrict__ idx_out) {
    __shared__ float sbest[WAVES][TQ][8][32];
    __shared__ int   sidx [WAVES][TQ][8][32];

    const int lane = threadIdx.x & 31;
    const int wave = threadIdx.x >> 5;
    const int qbase = blockIdx.x * QPB + wave * (TQ * 16);
    const int mrow = lane & 15;          // row (A) / col (B) within 16-tile
    const int koff = (lane >> 4) * 2;    // K sub-offset per lane half

    // Hoist A fragments: 3 tiles x 20 K-steps x v2f
    v2f a[TQ][20];
    #pragma unroll
    for (int t = 0; t < TQ; ++t) {
        const v2f* q2 = (const v2f*)(qp + (size_t)(qbase + t * 16 + mrow) * DP + koff);
        #pragma unroll
        for (int s = 0; s < 20; ++s) a[t][s] = q2[s * 2];
    }

    float best[TQ][8];
    int   bidx[TQ][8];
    #pragma unroll
    for (int t = 0; t < TQ; ++t)
        #pragma unroll
        for (int r = 0; r < 8; ++r) { best[t][r] = 3.4e38f; bidx[t][r] = 0; }

    for (int kb = 0; kb < NK; kb += 16) {
        const v2f* k2 = (const v2f*)(kp + (size_t)(kb + mrow) * DP + koff);
        v8f c0 = {}, c1 = {}, c2 = {};
        #pragma unroll
        for (int s = 0; s < 20; ++s) {
            v2f b = k2[s * 2];
            c0 = __builtin_amdgcn_wmma_f32_16x16x4_f32(false, a[0][s], false, b,
                                                       (short)0, c0, false, false);
            c1 = __builtin_amdgcn_wmma_f32_16x16x4_f32(false, a[1][s], false, b,
                                                       (short)0, c1, false, false);
            c2 = __builtin_amdgcn_wmma_f32_16x16x4_f32(false, a[2][s], false, b,
                                                       (short)0, c2, false, false);
        }
        const float knv = kn[kb + mrow];
        const int   key = kb + mrow;
        #pragma unroll
        for (int r = 0; r < 8; ++r) {
            float d0 = knv - 2.0f * c0[r];
            if (d0 < best[0][r]) { best[0][r] = d0; bidx[0][r] = key; }
            float d1 = knv - 2.0f * c1[r];
            if (d1 < best[1][r]) { best[1][r] = d1; bidx[1][r] = key; }
            float d2 = knv - 2.0f * c2[r];
            if (d2 < best[2][r]) { best[2][r] = d2; bidx[2][r] = key; }
        }
    }

    #pragma unroll
    for (int t = 0; t < TQ; ++t)
        #pragma unroll
        for (int r = 0; r < 8; ++r) {
            sbest[wave][t][r][lane] = best[t][r];
            sidx [wave][t][r][lane] = bidx[t][r];
        }
    __syncthreads();

    // One thread per query finalizes across the 16 lanes holding its row.
    for (int qloc = threadIdx.x; qloc < QPB; qloc += blockDim.x) {
        int w   = qloc / (TQ * 16);
        int rem = qloc - w * (TQ * 16);
        int t   = rem >> 4;
        int M   = rem & 15;
        int h   = M >> 3;       // which lane-half holds this row
        int r   = M & 7;        // which C VGPR
        float bb = 3.4e38f;
        int   bi = 0x7fffffff;
        for (int l = h * 16; l < h * 16 + 16; ++l) {
            float d = sbest[w][t][r][l];
            int   i = sidx [w][t][r][l];
            if (d < bb || (d == bb && i < bi)) { bb = d; bi = i; }
        }
        idx_out[blockIdx.x * QPB + qloc] = bi;
    }
}

// ---------------------------------------------------------------------------
// Deterministic sum((q - k[idx])^2): single block, fixed assignment,
// fixed-order LDS tree reduction. No float atomics.
// ---------------------------------------------------------------------------
__global__ __launch_bounds__(256) void loss_kernel(const float* __restrict__ qp,
                                                   const float* __restrict__ kp,
                                                   const int* __restrict__ idx,
                                                   float* __restrict__ out) {
    __shared__ float red[256];
    float acc = 0.0f;
    for (int n = threadIdx.x; n < NQ; n += 256) {
        const float* qr = qp + (size_t)n * DP;
        const float* kr = kp + (size_t)idx[n] * DP;
        #pragma unroll 5
        for (int j = 0; j < DR; ++j) {
            float d = qr[j] - kr[j];
            acc += d * d;
        }
    }
    red[threadIdx.x] = acc;
    __syncthreads();
    for (int s = 128; s > 0; s >>= 1) {
        if (threadIdx.x < s) red[threadIdx.x] += red[threadIdx.x + s];
        __syncthreads();
    }
    if (threadIdx.x == 0) out[0] = red[0];
}

// ---------------------------------------------------------------------------
extern "C" void kernel_launch(void* const* d_in, const int* in_sizes, int n_in,
                              void* d_out, int out_size, void* d_ws, size_t ws_size,
                              hipStream_t stream) {
    const float* result = (const float*)d_in[0];   // queries
    const float* valid  = (const float*)d_in[1];   // keys/values
    float* out = (float*)d_out;

    char* ws = (char*)d_ws;
    const size_t QBYTES = (size_t)NQ * DP * sizeof(float);   // 4,608,000
    float* qp = (float*)(ws);
    float* kp = (float*)(ws + QBYTES);
    float* kn = (float*)(ws + 2 * QBYTES);
    int*   ix = (int*)  (ws + 2 * QBYTES + (size_t)NK * sizeof(float));

    const int pe = NQ * DP;  // 1,152,000 -> 4500 blocks of 256
    patch_kernel<<<(pe + 255) / 256, 256, 0, stream>>>(result, qp);
    patch_kernel<<<(pe + 255) / 256, 256, 0, stream>>>(valid, kp);
    knorm_kernel<<<(NK + 255) / 256, 256, 0, stream>>>(kp, kn);
    argmin_kernel<<<NQ / QPB, 128, 0, stream>>>(qp, kp, kn, ix);   // 75 blocks
    loss_kernel<<<1, 256, 0, stream>>>(qp, kp, ix, out);
}